// GCN_air_75213467287806
// MI455X (gfx1250) — compile-verified
//
#include <hip/hip_runtime.h>
#include <hip/hip_bf16.h>
#include <math.h>

// GCNII forward for MI455X (gfx1250, wave32).
// fp32 GEMMs via V_WMMA_F32_16X16X4_F32 (full fp32 accuracy vs reference).
// Scatter-add over edges is the bandwidth-dominant phase (L2-resident agg).

#define HDIM 256
#define COUT 40
#define EPSV 1e-5f

typedef __attribute__((ext_vector_type(2))) float v2f;
typedef __attribute__((ext_vector_type(8))) float v8f;

static __device__ __forceinline__ v8f wmma_f32(v2f a, v2f b, v8f c) {
  // 8 args: (neg_a, A, neg_b, B, c_mod, C, reuse_a, reuse_b)
  return __builtin_amdgcn_wmma_f32_16x16x4_f32(
      false, a, false, b, (short)0, c, false, false);
}

__global__ __launch_bounds__(256) void k_fill_zero(float* p, size_t n) {
  size_t i = (size_t)blockIdx.x * blockDim.x + threadIdx.x;
  size_t st = (size_t)gridDim.x * blockDim.x;
  for (; i < n; i += st) p[i] = 0.0f;
}

__global__ __launch_bounds__(256) void k_deg(const int* __restrict__ row,
                                             float* deg, int E) {
  int e = blockIdx.x * blockDim.x + threadIdx.x;
  if (e < E) atomicAdd(&deg[row[e]], 1.0f);
}

__global__ __launch_bounds__(256) void k_dinv(float* deg, int n) {
  int i = blockIdx.x * blockDim.x + threadIdx.x;
  if (i < n) {
    float d = deg[i];
    deg[i] = (d > 0.0f) ? rsqrtf(fmaxf(d, 1.0f)) : 0.0f;
  }
}

// C = A[nrows x 256] * B[256 x 256]; optional fused (C + bias - mean)*scale + beta.
// One wave -> 16 rows x 64 cols (4 accumulators). nrows must be multiple of 16
// (100000 = 6250*16). Block = 256 threads = 8 waves = 128 rows.
__global__ __launch_bounds__(256)
void k_gemm_wmma(const float* __restrict__ A, const float* __restrict__ B,
                 float* __restrict__ O,
                 const float* __restrict__ bias, const float* __restrict__ gamma,
                 const float* __restrict__ beta, const float* __restrict__ mean,
                 const float* __restrict__ var, int nrows, int do_bn) {
  const int wave = threadIdx.x >> 5;
  const int lane = threadIdx.x & 31;
  const int rowbase = blockIdx.x * 128 + wave * 16;
  if (rowbase >= nrows) return;
  const int colbase = blockIdx.y * 64;
  const int m = lane & 15;   // row within A-tile / col within C-tile
  const int kh = lane >> 4;  // which K-pair (K=0,1 vs K=2,3)

  const float* arow = A + (size_t)(rowbase + m) * HDIM;
  v8f acc[4] = {};

  for (int kb = 0; kb < HDIM; kb += 4) {
    const int k0 = kb + 2 * kh;
    v2f a;
    a.x = arow[k0];
    a.y = arow[k0 + 1];
    const float* b0 = B + (size_t)k0 * HDIM + colbase + m;
    const float* b1 = b0 + HDIM;
#pragma unroll
    for (int t = 0; t < 4; t++) {
      v2f b;
      b.x = b0[16 * t];
      b.y = b1[16 * t];
      acc[t] = wmma_f32(a, b, acc[t]);
    }
  }

#pragma unroll
  for (int t = 0; t < 4; t++) {
    const int n = colbase + 16 * t + m;
    float sc = 1.0f, off = 0.0f;
    if (do_bn) {
      float s = gamma[n] * rsqrtf(var[n] + EPSV);
      sc = s;
      off = (bias[n] - mean[n]) * s + beta[n];
    }
#pragma unroll
    for (int j = 0; j < 8; j++) {
      int r = rowbase + j + 8 * kh;  // C/D layout: M = j + 8*(lane>>4)
      float val = acc[t][j];
      O[(size_t)r * HDIM + n] = do_bn ? (val * sc + off) : val;
    }
  }
}

// alpha = sigmoid([h, h_in] . Wp + bp); out = relu((1-alpha)*h + alpha*h_in)
// One wave per node; wave32 xor-shuffle reduction.
__global__ __launch_bounds__(256)
void k_alpha_mix(const float* __restrict__ h, const float* __restrict__ hin,
                 const float* __restrict__ Wp, const float* __restrict__ bp,
                 float* __restrict__ out, int n) {
  const int wave = threadIdx.x >> 5;
  const int lane = threadIdx.x & 31;
  const int node = blockIdx.x * 8 + wave;
  if (node >= n) return;
  const float* hr = h + (size_t)node * HDIM;
  const float* hr0 = hin + (size_t)node * HDIM;
  float s = 0.0f;
#pragma unroll
  for (int j = 0; j < 8; j++) {
    int idx = lane + 32 * j;
    s += hr[idx] * Wp[idx] + hr0[idx] * Wp[HDIM + idx];
  }
#pragma unroll
  for (int d = 16; d > 0; d >>= 1) s += __shfl_xor(s, d, 32);
  float alpha = 1.0f / (1.0f + expf(-(s + bp[0])));
  float* o = out + (size_t)node * HDIM;
#pragma unroll
  for (int j = 0; j < 8; j++) {
    int idx = lane + 32 * j;
    float v = (1.0f - alpha) * hr[idx] + alpha * hr0[idx];
    o[idx] = fmaxf(v, 0.0f);
  }
}

// agg[row[e]] += p[col[e]] * dinv[row[e]]*dinv[col[e]]; 64 float4-chunks/edge.
__global__ __launch_bounds__(256)
void k_scatter(const float* __restrict__ p, const int* __restrict__ row,
               const int* __restrict__ col, const float* __restrict__ dinv,
               float* __restrict__ agg, int E) {
  long gid = (long)blockIdx.x * blockDim.x + threadIdx.x;
  long total = (long)E * 64;
  if (gid >= total) return;
  int e = (int)(gid >> 6);
  int f = (int)(gid & 63) << 2;
  int r = row[e];
  int c = col[e];
  float w = dinv[r] * dinv[c];
  const float4 v = *(const float4*)(p + (size_t)c * HDIM + f);
  float* dst = agg + (size_t)r * HDIM + f;
  atomicAdd(dst + 0, v.x * w);
  atomicAdd(dst + 1, v.y * w);
  atomicAdd(dst + 2, v.z * w);
  atomicAdd(dst + 3, v.w * w);
}

// h = BN(h + bc) in place (eval-mode BN, bias folded in).
__global__ __launch_bounds__(256)
void k_bn_elem(float* __restrict__ h, const float* __restrict__ bcv,
               const float* __restrict__ g, const float* __restrict__ be,
               const float* __restrict__ mv, const float* __restrict__ vv,
               size_t total) {
  size_t i = (size_t)blockIdx.x * blockDim.x + threadIdx.x;
  size_t st = (size_t)gridDim.x * blockDim.x;
  for (; i < total; i += st) {
    int f = (int)(i & (HDIM - 1));
    float s = g[f] * rsqrtf(vv[f] + EPSV);
    h[i] = (h[i] + bcv[f] - mv[f]) * s + be[f];
  }
}

// out[node] = log_softmax(h[node] @ W_out + b_out); one 64-thread block per node.
__global__ __launch_bounds__(64)
void k_out_softmax(const float* __restrict__ h, const float* __restrict__ Wo,
                   const float* __restrict__ bo, float* __restrict__ out, int n) {
  int node = blockIdx.x;
  if (node >= n) return;
  int c = threadIdx.x;
  __shared__ float sv[COUT];
  __shared__ float stats[2];
  float acc = 0.0f;
  if (c < COUT) {
    acc = bo[c];
    const float* hr = h + (size_t)node * HDIM;
    for (int k = 0; k < HDIM; k++) acc += hr[k] * Wo[k * COUT + c];
    sv[c] = acc;
  }
  __syncthreads();
  if (c == 0) {
    float M = sv[0];
    for (int i = 1; i < COUT; i++) M = fmaxf(M, sv[i]);
    float S = 0.0f;
    for (int i = 0; i < COUT; i++) S += expf(sv[i] - M);
    stats[0] = M;
    stats[1] = logf(S);
  }
  __syncthreads();
  if (c < COUT) out[(size_t)node * COUT + c] = acc - stats[0] - stats[1];
}

extern "C" void kernel_launch(void* const* d_in, const int* in_sizes, int n_in,
                              void* d_out, int out_size, void* d_ws, size_t ws_size,
                              hipStream_t stream) {
  const float* x     = (const float*)d_in[0];
  const int*   row   = (const int*)d_in[1];
  const int*   col   = (const int*)d_in[2];
  const float* W_in  = (const float*)d_in[3];
  const float* b_in  = (const float*)d_in[4];
  const float* g0    = (const float*)d_in[5];
  const float* be0   = (const float*)d_in[6];
  const float* m0    = (const float*)d_in[7];
  const float* v0    = (const float*)d_in[8];
  const float* Wc    = (const float*)d_in[9];
  const float* bc    = (const float*)d_in[10];
  const float* gL    = (const float*)d_in[11];
  const float* beL   = (const float*)d_in[12];
  const float* mL    = (const float*)d_in[13];
  const float* vL    = (const float*)d_in[14];
  const float* Wp    = (const float*)d_in[15];
  const float* bp    = (const float*)d_in[16];
  const float* W_out = (const float*)d_in[17];
  const float* b_out = (const float*)d_in[18];

  const int N = in_sizes[0] / HDIM;
  const int E = in_sizes[1];
  const int L = in_sizes[9] / (HDIM * HDIM);

  float* ws = (float*)d_ws;
  size_t Npad = ((size_t)N + 255) & ~(size_t)255;
  size_t NH = (size_t)N * HDIM;
  float* dinv = ws;                 // [N]   degree -> d^-1/2 in place
  float* h    = ws + Npad;          // [N,H] current features / agg buffer
  float* hin  = h + NH;             // [N,H] initial residual h_in
  float* t1   = hin + NH;           // [N,H] gated/mixed features
  float* t2   = t1 + NH;            // [N,H] t1 @ Wc[i]

  // --- degrees -> dinv ---
  k_fill_zero<<<512, 256, 0, stream>>>(dinv, (size_t)N);
  k_deg<<<(E + 255) / 256, 256, 0, stream>>>(row, dinv, E);
  k_dinv<<<(N + 255) / 256, 256, 0, stream>>>(dinv, N);

  // --- input projection + BN: h = BN(x @ W_in + b_in) ---
  dim3 ggrid((N + 127) / 128, 4);
  k_gemm_wmma<<<ggrid, 256, 0, stream>>>(x, W_in, h, b_in, g0, be0, m0, v0, N, 1);
  hipMemcpyAsync(hin, h, NH * sizeof(float), hipMemcpyDeviceToDevice, stream);

  long sthreads = (long)E * 64;
  int sblocks = (int)((sthreads + 255) / 256);

  for (int i = 0; i < L; i++) {
    k_alpha_mix<<<(N + 7) / 8, 256, 0, stream>>>(h, hin, Wp, bp, t1, N);
    k_gemm_wmma<<<ggrid, 256, 0, stream>>>(t1, Wc + (size_t)i * HDIM * HDIM, t2,
                                           nullptr, nullptr, nullptr, nullptr,
                                           nullptr, N, 0);
    k_fill_zero<<<2048, 256, 0, stream>>>(h, NH);
    k_scatter<<<sblocks, 256, 0, stream>>>(t2, row, col, dinv, h, E);
    k_bn_elem<<<2048, 256, 0, stream>>>(h, bc + i * HDIM, gL + i * HDIM,
                                        beL + i * HDIM, mL + i * HDIM,
                                        vL + i * HDIM, NH);
  }

  // --- final gate + output projection + log_softmax ---
  k_alpha_mix<<<(N + 7) / 8, 256, 0, stream>>>(h, hin, Wp, bp, t1, N);
  k_out_softmax<<<N, 64, 0, stream>>>(t1, W_out, b_out, (float*)d_out, N);
}